// GraphAttention_23175643529906
// MI455X (gfx1250) — compile-verified
//
#include <hip/hip_runtime.h>
#include <hip/hip_bf16.h>

typedef __attribute__((ext_vector_type(2))) float v2f;
typedef __attribute__((ext_vector_type(4))) float v4f;
typedef __attribute__((ext_vector_type(8))) float v8f;

#define N_NODES 50000
#define N_EDGES 800000
#define DIM     128
#define NHEAD   8
#define DKH     16
#define NTILES  (N_NODES / 16)   // 3125 exactly
#define EH      (N_EDGES * NHEAD)
#define ND      (N_NODES * DIM)
#define NH      (N_NODES * NHEAD)

// ---- order-preserving float<->uint mapping for atomic segment-max ----
__device__ __forceinline__ unsigned fmap(float s) {
    unsigned u = __float_as_uint(s);
    return (u & 0x80000000u) ? ~u : (u | 0x80000000u);
}
__device__ __forceinline__ float funmap(unsigned u) {
    return (u & 0x80000000u) ? __uint_as_float(u & 0x7FFFFFFFu)
                             : __uint_as_float(~u);
}

// ---- pass 0: init accumulators ----
__global__ void k_init(float* __restrict__ att, unsigned* __restrict__ mmax,
                       float* __restrict__ denom) {
    int i = blockIdx.x * blockDim.x + threadIdx.x;
    if (i < ND) att[i] = 0.0f;
    if (i < NH) { mmax[i] = 0u; denom[i] = 0.0f; }  // mapped 0 == -inf identity
}

// ---- fused QKV projection: q/k/v = h @ W^T + b, fp32 WMMA 16x16x4 ----
__global__ __launch_bounds__(256) void k_qkv(
    const float* __restrict__ h,
    const float* __restrict__ Wq, const float* __restrict__ bq,
    const float* __restrict__ Wk, const float* __restrict__ bk,
    const float* __restrict__ Wv, const float* __restrict__ bv,
    float* __restrict__ qo, float* __restrict__ ko, float* __restrict__ vo) {
    int wave = blockIdx.x * (blockDim.x >> 5) + (threadIdx.x >> 5);
    if (wave >= NTILES) return;
    int lane = threadIdx.x & 31;
    int m15  = lane & 15;
    int g    = lane >> 4;

    // A fragments: 16x128 tile of h, held in registers, reused for all 24 tiles
    const float* hrow = h + (size_t)(wave * 16 + m15) * DIM;
    v2f aF[32];
#pragma unroll
    for (int kk = 0; kk < 32; ++kk)
        aF[kk] = *(const v2f*)(hrow + kk * 4 + g * 2);

    const float* Ws[3]   = {Wq, Wk, Wv};
    const float* bs[3]   = {bq, bk, bv};
    float*       outs[3] = {qo, ko, vo};

    for (int p = 0; p < 3; ++p) {
        const float* W = Ws[p];
        const float* b = bs[p];
        float*       o = outs[p];
        for (int nt = 0; nt < 8; ++nt) {
            int n = nt * 16 + m15;
            float bias = b[n];
            v8f acc;
#pragma unroll
            for (int r = 0; r < 8; ++r) acc[r] = bias;  // bias broadcast per column
            const float* Wrow = W + (size_t)n * DIM;    // B[k][n] = W[n][k]
#pragma unroll
            for (int kk = 0; kk < 32; ++kk) {
                v2f bF = *(const v2f*)(Wrow + kk * 4 + g * 2);
                acc = __builtin_amdgcn_wmma_f32_16x16x4_f32(
                    false, aF[kk], false, bF, (short)0, acc, false, false);
            }
            float* obase = o + (size_t)(wave * 16 + g * 8) * DIM + n;
#pragma unroll
            for (int r = 0; r < 8; ++r) obase[(size_t)r * DIM] = acc[r];
        }
    }
}

// ---- per-(edge,head) score + atomic segment-max ----
__global__ __launch_bounds__(256) void k_score(
    const float* __restrict__ q, const float* __restrict__ k,
    const int* __restrict__ src, const int* __restrict__ dst,
    float* __restrict__ score, unsigned* __restrict__ mmax) {
    int t = blockIdx.x * blockDim.x + threadIdx.x;
    if (t >= EH) return;
    int e = t >> 3, hd = t & 7;
    int d = dst[e], s = src[e];
    const v4f* qp = (const v4f*)(q + (size_t)d * DIM + hd * DKH);
    const v4f* kp = (const v4f*)(k + (size_t)s * DIM + hd * DKH);
    float acc = 0.0f;
#pragma unroll
    for (int i = 0; i < 4; ++i) {
        v4f a = qp[i], b = kp[i];
        acc += a.x * b.x + a.y * b.y + a.z * b.z + a.w * b.w;
    }
    acc *= 0.25f;  // 1/sqrt(DK)
    score[t] = acc;
    atomicMax(&mmax[d * NHEAD + hd], fmap(acc));
}

// ---- ex = exp(score - m[dst]) + atomic segment-sum of denom ----
__global__ __launch_bounds__(256) void k_exp(
    float* __restrict__ score, const unsigned* __restrict__ mmax,
    const int* __restrict__ dst, float* __restrict__ denom) {
    int t = blockIdx.x * blockDim.x + threadIdx.x;
    if (t >= EH) return;
    int e = t >> 3, hd = t & 7;
    int d = dst[e];
    float mm = funmap(mmax[d * NHEAD + hd]);
    float ex = __expf(score[t] - mm);
    score[t] = ex;  // in-place
    atomicAdd(&denom[d * NHEAD + hd], ex);
}

// ---- alpha * v[src] scatter-add into att[dst] ----
__global__ __launch_bounds__(256) void k_scatter(
    const float* __restrict__ ex, const float* __restrict__ denom,
    const float* __restrict__ v, const int* __restrict__ src,
    const int* __restrict__ dst, float* __restrict__ att) {
    int t = blockIdx.x * blockDim.x + threadIdx.x;
    if (t >= EH) return;
    int e = t >> 3, hd = t & 7;
    int d = dst[e], s = src[e];
    float alpha = ex[t] / denom[d * NHEAD + hd];
    const v4f* vp = (const v4f*)(v + (size_t)s * DIM + hd * DKH);
    float* ap = att + (size_t)d * DIM + hd * DKH;
#pragma unroll
    for (int i = 0; i < 4; ++i) {
        v4f vv = vp[i];
        atomicAdd(ap + i * 4 + 0, alpha * vv.x);
        atomicAdd(ap + i * 4 + 1, alpha * vv.y);
        atomicAdd(ap + i * 4 + 2, alpha * vv.z);
        atomicAdd(ap + i * 4 + 3, alpha * vv.w);
    }
}

// ---- output projection: out = att @ Wo^T + bo, fp32 WMMA ----
__global__ __launch_bounds__(256) void k_proj(
    const float* __restrict__ X, const float* __restrict__ W,
    const float* __restrict__ b, float* __restrict__ out) {
    int wave = blockIdx.x * (blockDim.x >> 5) + (threadIdx.x >> 5);
    if (wave >= NTILES) return;
    int lane = threadIdx.x & 31;
    int m15  = lane & 15;
    int g    = lane >> 4;

    const float* xrow = X + (size_t)(wave * 16 + m15) * DIM;
    v2f aF[32];
#pragma unroll
    for (int kk = 0; kk < 32; ++kk)
        aF[kk] = *(const v2f*)(xrow + kk * 4 + g * 2);

    for (int nt = 0; nt < 8; ++nt) {
        int n = nt * 16 + m15;
        float bias = b[n];
        v8f acc;
#pragma unroll
        for (int r = 0; r < 8; ++r) acc[r] = bias;
        const float* Wrow = W + (size_t)n * DIM;
#pragma unroll
        for (int kk = 0; kk < 32; ++kk) {
            v2f bF = *(const v2f*)(Wrow + kk * 4 + g * 2);
            acc = __builtin_amdgcn_wmma_f32_16x16x4_f32(
                false, aF[kk], false, bF, (short)0, acc, false, false);
        }
        float* obase = out + (size_t)(wave * 16 + g * 8) * DIM + n;
#pragma unroll
        for (int r = 0; r < 8; ++r) obase[(size_t)r * DIM] = acc[r];
    }
}

extern "C" void kernel_launch(void* const* d_in, const int* in_sizes, int n_in,
                              void* d_out, int out_size, void* d_ws, size_t ws_size,
                              hipStream_t stream) {
    const float* h  = (const float*)d_in[0];
    const float* Wq = (const float*)d_in[1];
    const float* bq = (const float*)d_in[2];
    const float* Wk = (const float*)d_in[3];
    const float* bk = (const float*)d_in[4];
    const float* Wv = (const float*)d_in[5];
    const float* bv = (const float*)d_in[6];
    const float* Wo = (const float*)d_in[7];
    const float* bo = (const float*)d_in[8];
    const int*  src = (const int*)d_in[9];
    const int*  dst = (const int*)d_in[10];
    float* out = (float*)d_out;

    // workspace layout (floats): q | k | v | att | score/ex | mmax(u32) | denom
    float* ws = (float*)d_ws;
    float*    q_ws     = ws;
    float*    k_ws     = q_ws + (size_t)ND;
    float*    v_ws     = k_ws + (size_t)ND;
    float*    att_ws   = v_ws + (size_t)ND;
    float*    score_ws = att_ws + (size_t)ND;
    unsigned* m_ws     = (unsigned*)(score_ws + (size_t)EH);
    float*    den_ws   = (float*)(m_ws + (size_t)NH);

    const int BLK = 256;
    int gInit  = (ND + BLK - 1) / BLK;                 // covers ND >= NH
    int gEdge  = (EH + BLK - 1) / BLK;
    int gWmma  = (NTILES + (BLK / 32) - 1) / (BLK / 32);

    k_init<<<gInit, BLK, 0, stream>>>(att_ws, m_ws, den_ws);
    k_qkv<<<gWmma, BLK, 0, stream>>>(h, Wq, bq, Wk, bk, Wv, bv, q_ws, k_ws, v_ws);
    k_score<<<gEdge, BLK, 0, stream>>>(q_ws, k_ws, src, dst, score_ws, m_ws);
    k_exp<<<gEdge, BLK, 0, stream>>>(score_ws, m_ws, dst, den_ws);
    k_scatter<<<gEdge, BLK, 0, stream>>>(score_ws, den_ws, v_ws, src, dst, att_ws);
    k_proj<<<gWmma, BLK, 0, stream>>>(att_ws, Wo, bo, out);
}